// FusionBlock_46127948759313
// MI455X (gfx1250) — compile-verified
//
#include <hip/hip_runtime.h>
#include <hip/hip_bf16.h>
#include <stdint.h>

// Problem constants (fixed by the reference):
#define NROW 4096   // n
#define DDIM 256    // D
#define BATCH 4     // B
#define TOPK 20     // k

#define NEG_BIG (-3.402823466e38f)

typedef __attribute__((ext_vector_type(2))) float v2f;
typedef __attribute__((ext_vector_type(8))) float v8f;

// -----------------------------------------------------------------------------
// Kernel 1: per-row softmax statistics + iterative top-20 selection.
// One workgroup (256 thr = 8 wave32) per row of the 4096x4096 memory matrix.
// Row is staged into LDS via CDNA5 async global->LDS DMA (ASYNCcnt).
// -----------------------------------------------------------------------------
__global__ __launch_bounds__(256) void topk_softmax_kernel(
    const float* __restrict__ mem, float* __restrict__ wv, int* __restrict__ wi)
{
    __shared__ float s_row[NROW];   // 16 KB row cache
    __shared__ float s_rv[8];
    __shared__ int   s_ri[8];
    __shared__ float s_par[2];      // [0]=row max, [1]=1/sum(exp)

    const int row  = blockIdx.x;
    const int tid  = threadIdx.x;
    const int lane = tid & 31;
    const int wid  = tid >> 5;
    const float* rp = mem + (size_t)row * NROW;

    // --- async copy row -> LDS (GLOBAL_LOAD_ASYNC_TO_LDS_B128, GVS mode) ---
    {
        uint32_t lds_base = (uint32_t)(uintptr_t)&s_row[0]; // low 32 bits = LDS offset
        for (int t = tid; t < NROW / 4; t += 256) {
            uint32_t la = lds_base + (uint32_t)t * 16u;  // LDS dest (bytes)
            uint32_t go = (uint32_t)t * 16u;             // global byte offset
            asm volatile("global_load_async_to_lds_b128 %0, %1, %2 offset:0"
                         :: "v"(la), "v"(go), "s"(rp) : "memory");
        }
        asm volatile("s_wait_asynccnt 0" ::: "memory");
    }
    __syncthreads();

    // --- row max (wave32 shuffle reduce, then cross-wave via LDS) ---
    float m = NEG_BIG;
    for (int i = tid; i < NROW; i += 256) m = fmaxf(m, s_row[i]);
    for (int off = 16; off > 0; off >>= 1) m = fmaxf(m, __shfl_xor(m, off, 32));
    if (lane == 0) s_rv[wid] = m;
    __syncthreads();
    if (tid == 0) {
        float mm = s_rv[0];
#pragma unroll
        for (int i = 1; i < 8; ++i) mm = fmaxf(mm, s_rv[i]);
        s_par[0] = mm;
    }
    __syncthreads();
    const float rmax = s_par[0];

    // --- sum of exp(x - max) ---
    float ss = 0.f;
    for (int i = tid; i < NROW; i += 256) ss += __expf(s_row[i] - rmax);
    for (int off = 16; off > 0; off >>= 1) ss += __shfl_xor(ss, off, 32);
    if (lane == 0) s_rv[wid] = ss;
    __syncthreads();
    if (tid == 0) {
        float t = 0.f;
#pragma unroll
        for (int i = 0; i < 8; ++i) t += s_rv[i];
        s_par[1] = 1.0f / t;
    }
    __syncthreads();
    const float invz = s_par[1];

    // --- 20 iterative argmax passes over the LDS-resident row ---
    // softmax is monotone => top-k of logits == top-k of probs.
    for (int k = 0; k < TOPK; ++k) {
        float bv = NEG_BIG;
        int   bi = 0x7fffffff;
        for (int i = tid; i < NROW; i += 256) {
            float v = s_row[i];
            if (v > bv || (v == bv && i < bi)) { bv = v; bi = i; }
        }
        for (int off = 16; off > 0; off >>= 1) {
            float ov = __shfl_xor(bv, off, 32);
            int   oi = __shfl_xor(bi, off, 32);
            if (ov > bv || (ov == bv && oi < bi)) { bv = ov; bi = oi; }
        }
        if (lane == 0) { s_rv[wid] = bv; s_ri[wid] = bi; }
        __syncthreads();
        if (tid == 0) {
            float fv = s_rv[0]; int fi = s_ri[0];
#pragma unroll
            for (int i = 1; i < 8; ++i) {
                if (s_rv[i] > fv || (s_rv[i] == fv && s_ri[i] < fi)) {
                    fv = s_rv[i]; fi = s_ri[i];
                }
            }
            wv[(size_t)row * TOPK + k] = __expf(fv - rmax) * invz;
            wi[(size_t)row * TOPK + k] = fi;
            s_row[fi] = NEG_BIG;   // exclude from later passes
        }
        __syncthreads();
    }
}

// -----------------------------------------------------------------------------
// Kernel 2: gather 20 src2 rows per (row,batch) into LDS and compute
// out[b,row,:] = src1[b,row,:] + sum_k w[row,k] * src2[b,idx[row,k],:]
// using V_WMMA_F32_16X16X4_F32 (exact fp32 MMA):
//   A[16x4] = gathered src2 columns (M = d within 16-wide tile, K = k-chunk)
//   B[4x16] = top-k weight broadcast across N
//   C[m][n] = out[d0+m] replicated across n; extract column N=0.
// Grid: (NROW, BATCH), 256 threads (8 waves; each wave owns 2 d-tiles of 16).
// -----------------------------------------------------------------------------
__global__ __launch_bounds__(256) void gather_wsum_wmma_kernel(
    const float* __restrict__ src1, const float* __restrict__ src2,
    const float* __restrict__ wv, const int* __restrict__ wi,
    float* __restrict__ out)
{
    __shared__ float s_g[TOPK * DDIM];  // 20 KB gathered rows
    __shared__ float s_w[TOPK];
    __shared__ int   s_i[TOPK];
    __shared__ float s_out[DDIM];

    const int row = blockIdx.x;
    const int b   = blockIdx.y;
    const int tid = threadIdx.x;
    const int lane = tid & 31;
    const int wid  = tid >> 5;

    if (tid < TOPK) {
        s_w[tid] = wv[(size_t)row * TOPK + tid];
        s_i[tid] = wi[(size_t)row * TOPK + tid];
    }
    __syncthreads();

    // --- gather 20 rows of src2[b] into LDS (float4) ---
    for (int t = tid; t < TOPK * (DDIM / 4); t += 256) {
        int k  = t >> 6;          // DDIM/4 == 64
        int d4 = t & 63;
        const float4* srow =
            (const float4*)(src2 + ((size_t)b * NROW + (size_t)s_i[k]) * DDIM);
        ((float4*)s_g)[t] = srow[d4];
    }
    __syncthreads();

    // --- WMMA weighted sum: each wave computes 2 tiles of 16 d-values ---
#pragma unroll
    for (int tt = 0; tt < 2; ++tt) {
        const int tile = wid * 2 + tt;
        const int d0   = tile * 16;
        v8f acc = {};
#pragma unroll
        for (int c = 0; c < TOPK / 4; ++c) {     // 5 K-chunks of 4
            // A (16x4 f32): lanes 0-15 -> K = 4c+0 (v0), 4c+1 (v1);
            //               lanes 16-31 -> K = 4c+2 (v0), 4c+3 (v1)
            const int m  = lane & 15;
            const int kb = c * 4 + ((lane >> 4) << 1);
            v2f A;
            A.x = s_g[(kb + 0) * DDIM + d0 + m];
            A.y = s_g[(kb + 1) * DDIM + d0 + m];
            // B (4x16 f32): row = weight broadcast over N
            v2f Bm;
            Bm.x = s_w[c * 4 + (lane >> 4)];
            Bm.y = s_w[c * 4 + 2 + (lane >> 4)];
            acc = __builtin_amdgcn_wmma_f32_16x16x4_f32(
                /*neg_a=*/false, A, /*neg_b=*/false, Bm,
                /*c_mod=*/(short)0, acc, /*reuse_a=*/false, /*reuse_b=*/false);
        }
        // C layout: VGPR r -> (M=r, N=lane) lanes 0-15 ; (M=r+8, N=lane-16) lanes 16-31.
        // All N columns identical; take N==0 (lanes 0 and 16).
        if ((lane & 15) == 0) {
            const int mb = (lane >> 4) * 8;
#pragma unroll
            for (int r = 0; r < 8; ++r) s_out[d0 + mb + r] = acc[r];
        }
    }
    __syncthreads();

    // --- epilogue: add src1 and store ---
    const size_t base = ((size_t)b * NROW + (size_t)row) * DDIM;
    for (int d = tid; d < DDIM; d += 256) out[base + d] = src1[base + d] + s_out[d];
}

// -----------------------------------------------------------------------------
extern "C" void kernel_launch(void* const* d_in, const int* in_sizes, int n_in,
                              void* d_out, int out_size, void* d_ws, size_t ws_size,
                              hipStream_t stream)
{
    const float* src1 = (const float*)d_in[0];   // [4,4096,256] f32
    const float* src2 = (const float*)d_in[1];   // [4,4096,256] f32
    const float* memM = (const float*)d_in[2];   // [4096,4096] f32
    float* out = (float*)d_out;                  // [4,4096,256] f32

    // workspace: top-k weights then indices
    float* wv = (float*)d_ws;                                          // n*20 f32
    int*   wi = (int*)((char*)d_ws + (size_t)NROW * TOPK * sizeof(float)); // n*20 i32

    topk_softmax_kernel<<<NROW, 256, 0, stream>>>(memM, wv, wi);

    dim3 g2(NROW, BATCH);
    gather_wsum_wmma_kernel<<<g2, 256, 0, stream>>>(src1, src2, wv, wi, out);
}